// GCNConvLayer_22084721836888
// MI455X (gfx1250) — compile-verified
//
#include <hip/hip_runtime.h>

// CDNA5 / gfx1250 WMMA vector types
typedef __attribute__((ext_vector_type(2))) float v2f;
typedef __attribute__((ext_vector_type(8))) float v8f;

#define D_DIM 128

// ---------------------------------------------------------------------------
// Kernel 1: init accumulator to 0, degrees to 1.0 (the "+1" normalization)
// ---------------------------------------------------------------------------
__global__ void gcn_init_kernel(float* __restrict__ neigh, float* __restrict__ degs,
                                int ND, int N) {
    int i = blockIdx.x * blockDim.x + threadIdx.x;
    if (i < ND) neigh[i] = 0.0f;
    if (i < N)  degs[i]  = 1.0f;
}

// ---------------------------------------------------------------------------
// Kernel 2: per-edge  m = nfeat[src] + sum_f edge_emb[f][idx_f]  ->
//           atomic scatter-add onto neigh[dst], degree count onto degs[dst].
// 32 lanes per edge, each lane owns 4 consecutive floats (float4 gathers).
// ---------------------------------------------------------------------------
__global__ __launch_bounds__(256) void gcn_edge_kernel(
    const float* __restrict__ nfeat,
    const int*   __restrict__ src,
    const int*   __restrict__ dst,
    const int*   __restrict__ eidx,   // [E,3]
    const float* __restrict__ emb,    // [3,8,128]
    float* __restrict__ neigh,
    float* __restrict__ degs,
    int E)
{
    int e = blockIdx.x * 8 + (threadIdx.x >> 5);
    if (e >= E) return;
    int lane = threadIdx.x & 31;

    int s  = src[e];
    int d  = dst[e];
    int i0 = eidx[e * 3 + 0];
    int i1 = eidx[e * 3 + 1];
    int i2 = eidx[e * 3 + 2];

    const float4* nf = (const float4*)(nfeat + (size_t)s * D_DIM);
    const float4* t0 = (const float4*)(emb + (size_t)(0 * 8 + i0) * D_DIM);
    const float4* t1 = (const float4*)(emb + (size_t)(1 * 8 + i1) * D_DIM);
    const float4* t2 = (const float4*)(emb + (size_t)(2 * 8 + i2) * D_DIM);

    float4 a  = nf[lane];
    float4 b0 = t0[lane];
    float4 b1 = t1[lane];
    float4 b2 = t2[lane];

    float m0 = a.x + b0.x + b1.x + b2.x;
    float m1 = a.y + b0.y + b1.y + b2.y;
    float m2 = a.z + b0.z + b1.z + b2.z;
    float m3 = a.w + b0.w + b1.w + b2.w;

    float* o = neigh + (size_t)d * D_DIM + lane * 4;
    unsafeAtomicAdd(o + 0, m0);   // -> global_atomic_add_f32 (no CAS loop)
    unsafeAtomicAdd(o + 1, m1);
    unsafeAtomicAdd(o + 2, m2);
    unsafeAtomicAdd(o + 3, m3);
    if (lane == 0) unsafeAtomicAdd(&degs[d], 1.0f);
}

// ---------------------------------------------------------------------------
// Kernel 3: X = (nfeat + neigh) / degs   (written in-place over neigh buffer)
// ---------------------------------------------------------------------------
__global__ __launch_bounds__(256) void gcn_finalize_kernel(
    const float* __restrict__ nfeat,
    float* __restrict__ neighX,
    const float* __restrict__ degs,
    int N)
{
    int i = blockIdx.x * blockDim.x + threadIdx.x;   // over N*32 float4s
    if (i >= N * (D_DIM / 4)) return;
    int row = i >> 5;                                 // 32 float4s per row
    float inv = 1.0f / degs[row];
    float4 nf = ((const float4*)nfeat)[i];
    float4 ng = ((float4*)neighX)[i];
    float4 x;
    x.x = (nf.x + ng.x) * inv;
    x.y = (nf.y + ng.y) * inv;
    x.z = (nf.z + ng.z) * inv;
    x.w = (nf.w + ng.w) * inv;
    ((float4*)neighX)[i] = x;
}

// ---------------------------------------------------------------------------
// Kernel 4: out = X @ W + b  using V_WMMA_F32_16X16X4_F32 (full fp32).
// Block = 256 threads = 8 waves. Block owns 16 rows of X (staged in LDS,
// stride padded to 130 floats to avoid bank conflicts). Wave w owns the
// 16-column tile [16w, 16w+16). K loop steps by 4.
//
// ISA layouts (CDNA5 ISA):
//   A 16x4 f32:  lane%16 = M,  K = vgpr + 2*(lane/16)
//   B 4x16 f32:  lane%16 = N,  K = vgpr + 2*(lane/16)
//   C 16x16 f32: vgpr v, lanes 0-15: M=v, N=lane; lanes 16-31: M=v+8
//
// Stores: uniform full-tile fast path (single base pointer + immediate
// offsets, no per-lane exec masking) since rowBase+16 <= N for all blocks
// when N % 16 == 0.
// ---------------------------------------------------------------------------
__global__ __launch_bounds__(256) void gcn_gemm_wmma_kernel(
    const float* __restrict__ X,     // [N,128]
    const float* __restrict__ W,     // [128,128]
    const float* __restrict__ bias,  // [128]
    float* __restrict__ out,         // [N,128]
    int N)
{
    __shared__ float Xs[16 * 130];

    const int rowBase = blockIdx.x * 16;
    const int tid = threadIdx.x;

    // Stage 16x128 block of X into LDS (each thread: 2 float4 loads).
    #pragma unroll
    for (int j = tid; j < 512; j += 256) {
        int r  = j >> 5;        // 0..15
        int c4 = j & 31;        // 0..31
        int gr = rowBase + r;
        if (gr >= N) gr = N - 1;                       // clamp, keep EXEC full
        float4 v = ((const float4*)(X + (size_t)gr * D_DIM))[c4];
        float* p = &Xs[r * 130 + c4 * 4];
        ((float2*)p)[0] = make_float2(v.x, v.y);       // 8B-aligned DS stores
        ((float2*)p)[1] = make_float2(v.z, v.w);
    }
    __syncthreads();

    const int wave    = tid >> 5;
    const int lane    = tid & 31;
    const int colBase = wave * 16;
    const int lmod    = lane & 15;
    const int lhi     = lane >> 4;          // 0 or 1

    v8f c = {0.0f, 0.0f, 0.0f, 0.0f, 0.0f, 0.0f, 0.0f, 0.0f};
    const float* Arow = &Xs[lmod * 130];

    #pragma unroll 4
    for (int k0 = 0; k0 < D_DIM; k0 += 4) {
        int k = k0 + 2 * lhi;
        v2f a;
        a.x = Arow[k];
        a.y = Arow[k + 1];
        v2f b;
        b.x = W[(size_t)k       * D_DIM + colBase + lmod];
        b.y = W[(size_t)(k + 1) * D_DIM + colBase + lmod];
        // (neg_a, A, neg_b, B, c_mod, C, reuse_a, reuse_b)
        c = __builtin_amdgcn_wmma_f32_16x16x4_f32(false, a, false, b,
                                                  (short)0, c, false, false);
    }

    float bv = bias[colBase + lmod];

    // Base pointer for this lane's column: rows rowBase+8*lhi .. +7 stride 128.
    float* obase = out + (size_t)(rowBase + 8 * lhi) * D_DIM + colBase + lmod;

    if (rowBase + 16 <= N) {
        // Uniform fast path: 8 stores at constant immediate offsets.
        #pragma unroll
        for (int v = 0; v < 8; ++v)
            obase[(size_t)v * D_DIM] = c[v] + bv;
    } else {
        #pragma unroll
        for (int v = 0; v < 8; ++v) {
            int row = rowBase + 8 * lhi + v;
            if (row < N)
                obase[(size_t)v * D_DIM] = c[v] + bv;
        }
    }
}

// ---------------------------------------------------------------------------
// Host-side launcher
// Inputs (setup_inputs order):
//   0 nfeat [N,128] f32 | 1 src [E] i32 | 2 dst [E] i32 | 3 efeat_idx [E,3] i32
//   4 edge_emb [3,8,128] f32 | 5 W [128,128] f32 | 6 b [128] f32
// Workspace: neigh/X accumulator (N*128 f32) + degs (N f32)  ~= 25.8 MB
// ---------------------------------------------------------------------------
extern "C" void kernel_launch(void* const* d_in, const int* in_sizes, int n_in,
                              void* d_out, int out_size, void* d_ws, size_t ws_size,
                              hipStream_t stream) {
    const float* nfeat = (const float*)d_in[0];
    const int*   src   = (const int*)d_in[1];
    const int*   dst   = (const int*)d_in[2];
    const int*   eidx  = (const int*)d_in[3];
    const float* emb   = (const float*)d_in[4];
    const float* W     = (const float*)d_in[5];
    const float* bias  = (const float*)d_in[6];
    float*       out   = (float*)d_out;

    const int N  = in_sizes[0] / D_DIM;
    const int E  = in_sizes[1];
    const int ND = N * D_DIM;

    float* neighX = (float*)d_ws;
    float* degs   = neighX + (size_t)ND;

    gcn_init_kernel<<<(ND + 255) / 256, 256, 0, stream>>>(neighX, degs, ND, N);
    gcn_edge_kernel<<<(E + 7) / 8, 256, 0, stream>>>(nfeat, src, dst, eidx, emb,
                                                     neighX, degs, E);
    gcn_finalize_kernel<<<(ND / 4 + 255) / 256, 256, 0, stream>>>(nfeat, neighX,
                                                                  degs, N);
    gcn_gemm_wmma_kernel<<<(N + 15) / 16, 256, 0, stream>>>(neighX, W, bias, out, N);
}